// PostProcessPooling_53395033423883
// MI455X (gfx1250) — compile-verified
//
#include <hip/hip_runtime.h>
#include <hip/hip_bf16.h>
#include <stdint.h>

// ---------------------------------------------------------------------------
// PostProcessPooling for MI455X (gfx1250), compile-only tuning loop.
//
// Phase 1: threshold 200K chord scores                 (trivial, streaming)
// Phase 2: sequential greedy clustering of 200K edges  (latency bound)
//          -> whole cluster[40000] (156KB) + mask bitset live in the WGP's
//             320KB LDS (CDNA5-only; CDNA4's 64KB cannot hold it), and the
//             chord-edge stream is double-buffered into LDS by the Tensor
//             Data Mover (tensor_load_to_lds + s_wait_tensorcnt).
// Phase 3: coalesce 8M edges by (cluster[src],cluster[dst]) with mean probs
//          -> L2-resident open-addressed hash (192MB L2) + compaction.
//             NOTE: emission order of unique pairs differs from the
//             reference's sorted jnp.unique order; per-pair means match.
// ---------------------------------------------------------------------------

#define THRESH    0.5f
#define MAXN      40960            // LDS-resident cluster-state capacity
#define CH        2048             // chord staging chunk (elements)
#define EMPTY_KEY (-2147483647 - 1)

typedef unsigned int v4u __attribute__((ext_vector_type(4)));
typedef int          v8i __attribute__((ext_vector_type(8)));
typedef int          v4i __attribute__((ext_vector_type(4)));

#if __has_builtin(__builtin_amdgcn_tensor_load_to_lds) && \
    __has_builtin(__builtin_amdgcn_s_wait_tensorcnt)
#define USE_TDM 1
#else
#define USE_TDM 0
#endif

// 1-D TDM descriptor: tile of `len` 4-byte elements, global -> LDS.
// This toolchain exposes the 6-arg builtin:
//   (uint32x4 g0, int32x8 g1, int32x4 g2, int32x4 g3, int32x8 g4, i32 cpol)
__device__ __forceinline__ void tdm_load_1d(unsigned lds_off, const void* gsrc, int len) {
#if USE_TDM
  unsigned long long ga = (unsigned long long)(uintptr_t)gsrc;
  v4u g0;
  g0.x = 1u;                                                  // count=1 (user D#)
  g0.y = lds_off;                                             // lds_addr
  g0.z = (unsigned)(ga & 0xFFFFFFFFull);                      // global_addr[31:0]
  g0.w = (unsigned)((ga >> 32) & 0x1FFFFFFull) | (2u << 30);  // addr[56:32] | type=2
  v8i g1;
  g1[0] = 2 << 16;                              // data_size=4B, no multicast/pad
  g1[1] = (len & 0xFFFF) << 16;                 // tensor_dim0[15:0]
  g1[2] = ((len >> 16) & 0xFFFF) | (1 << 16);   // tensor_dim0[31:16] | tensor_dim1=1
  g1[3] = (len & 0xFFFF) << 16;                 // tile_dim0=len
  g1[4] = 0;                                    // tile_dim1=tile_dim2=0 (1-D)
  g1[5] = len;                                  // tensor_dim0_stride
  g1[6] = 0;
  g1[7] = 0;
  v4i g2 = {0, 0, 0, 0};
  v4i g3 = {0, 0, 0, 0};
  v8i g4 = {0, 0, 0, 0, 0, 0, 0, 0};
  __builtin_amdgcn_tensor_load_to_lds(g0, g1, g2, g3, g4, 0);
#else
  (void)lds_off; (void)gsrc; (void)len;
#endif
}

// ---------------- Phase 1: edge_score = (score > 0.5) ----------------------
__global__ void k_score(const float* __restrict__ scr, float* __restrict__ o_score, int n) {
  int i = blockIdx.x * blockDim.x + threadIdx.x;
  if (i < n) o_score[i] = (scr[i] > THRESH) ? 1.0f : 0.0f;
}

// ---------------- init outputs / hash / misc -------------------------------
__global__ void k_init(float* __restrict__ o_row, float* __restrict__ o_col,
                       float* __restrict__ o_prob, int E,
                       int* __restrict__ hkeys, float* __restrict__ hsum,
                       int* __restrict__ hcnt, int slots,
                       int* __restrict__ ws_nb, int n, int* __restrict__ counter) {
  long long i = (long long)blockIdx.x * blockDim.x + threadIdx.x;
  if (i < E)     { o_row[i] = -1.0f; o_col[i] = -1.0f; o_prob[i] = 0.0f; }
  if (i < slots) { hkeys[i] = EMPTY_KEY; hsum[i] = 0.0f; hcnt[i] = 0; }
  if (i < n)     { ws_nb[i] = 0; }
  if (i == 0)    { counter[0] = 0; }
}

// ---------------- Phase 2: sequential greedy clustering --------------------
// Single wave32. cluster[] + mask bitset LDS-resident; chord stream DMA'd in
// by the TDM, double buffered, consumed serially by lane 0.
__global__ void __launch_bounds__(32)
k_greedy(const int* __restrict__ csrc, const int* __restrict__ cdst,
         const float* __restrict__ cscr,
         int* __restrict__ ws_cluster,
         float* __restrict__ o_cluster, float* __restrict__ o_reduced,
         int n_nodes, int n_chord) {
  __shared__ int      lds_cluster[MAXN];       // 160.0 KB
  __shared__ unsigned lds_mask[MAXN / 32];     //   5.0 KB
  __shared__ int      st_src[2][CH];           //  16.0 KB
  __shared__ int      st_dst[2][CH];           //  16.0 KB
  __shared__ float    st_scr[2][CH];           //  16.0 KB   (total ~213 KB < 320 KB)
  __shared__ int      sh_reduced;

  const int lane = threadIdx.x;
  for (int i = lane; i < n_nodes; i += 32)   lds_cluster[i] = 0;
  for (int i = lane; i < MAXN / 32; i += 32) lds_mask[i] = 0xFFFFFFFFu;
  __syncthreads();

  const int nch = (n_chord + CH - 1) / CH;
  int idx = 0;  // only lane 0's copy is meaningful

#if USE_TDM
  {
    const int len0 = min(CH, n_chord);
    tdm_load_1d((unsigned)(uintptr_t)(void*)&st_src[0][0], csrc, len0);
    tdm_load_1d((unsigned)(uintptr_t)(void*)&st_dst[0][0], cdst, len0);
    tdm_load_1d((unsigned)(uintptr_t)(void*)&st_scr[0][0], cscr, len0);
  }
#endif

  for (int c = 0; c < nch; ++c) {
    const int buf = c & 1;
    const int len = min(CH, n_chord - c * CH);
#if USE_TDM
    if (c + 1 < nch) {                       // prefetch next chunk, then wait
      const int nb = buf ^ 1;                // for current one (TDM in-order)
      const int nl = min(CH, n_chord - (c + 1) * CH);
      tdm_load_1d((unsigned)(uintptr_t)(void*)&st_src[nb][0], csrc + (c + 1) * CH, nl);
      tdm_load_1d((unsigned)(uintptr_t)(void*)&st_dst[nb][0], cdst + (c + 1) * CH, nl);
      tdm_load_1d((unsigned)(uintptr_t)(void*)&st_scr[nb][0], cscr + (c + 1) * CH, nl);
      __builtin_amdgcn_s_wait_tensorcnt(3);
    } else {
      __builtin_amdgcn_s_wait_tensorcnt(0);
    }
#else
    for (int j = lane; j < len; j += 32) {
      st_src[buf][j] = csrc[c * CH + j];
      st_dst[buf][j] = cdst[c * CH + j];
      st_scr[buf][j] = cscr[c * CH + j];
    }
    __syncthreads();
#endif
    if (lane == 0) {
      for (int j = 0; j < len; ++j) {
        const int s = st_src[buf][j];
        const int d = st_dst[buf][j];
        if (st_scr[buf][j] > THRESH) {
          const unsigned bs = 1u << (s & 31), bd = 1u << (d & 31);
          const bool ms = (lds_mask[s >> 5] & bs) != 0;
          const bool md = (lds_mask[d >> 5] & bd) != 0;
          if (ms && md) {                          // new cluster (self-loop OK)
            lds_cluster[s] = idx; lds_cluster[d] = idx;
            lds_mask[s >> 5] &= ~bs; lds_mask[d >> 5] &= ~bd;
            ++idx;
          } else if (ms) {                         // join dst's cluster
            lds_cluster[s] = lds_cluster[d];
            lds_mask[s >> 5] &= ~bs;
          } else if (md) {                         // join src's cluster
            lds_cluster[d] = lds_cluster[s];
            lds_mask[d >> 5] &= ~bd;
          }
        }
      }
    }
    __syncthreads();
  }

  // remaining masked nodes -> singleton clusters in node order
  if (lane == 0) {
    int run = 0;
    for (int i = 0; i < n_nodes; ++i) {
      if ((lds_mask[i >> 5] >> (i & 31)) & 1u) { lds_cluster[i] = idx + run; ++run; }
    }
    sh_reduced = idx + run;
  }
  __syncthreads();

  for (int i = lane; i < n_nodes; i += 32) {
    const int cv = lds_cluster[i];
    ws_cluster[i] = cv;
    o_cluster[i]  = (float)cv;
  }
  if (lane == 0) o_reduced[0] = (float)sh_reduced;
}

// ---------------- new_batch scatter ----------------------------------------
__global__ void k_newbatch(const int* __restrict__ cluster, const int* __restrict__ batch,
                           int* __restrict__ ws_nb, int n) {
  int i = blockIdx.x * blockDim.x + threadIdx.x;
  if (i < n) atomicMax(&ws_nb[cluster[i]], batch[i]);  // deterministic dup resolve
}

// ---------------- Phase 3a: hash accumulate (sum, count) per pair-key ------
__global__ void k_coalesce(const int* __restrict__ e0, const int* __restrict__ e1,
                           const float* __restrict__ probs,
                           const int* __restrict__ cluster,
                           int* __restrict__ hkeys, float* __restrict__ hsum,
                           int* __restrict__ hcnt,
                           int E, int n_nodes, unsigned smask) {
  const int stride = gridDim.x * blockDim.x;
  for (int i = blockIdx.x * blockDim.x + threadIdx.x; i < E; i += stride) {
    if (i + stride < E) {                       // warm next grid-stride slice
      __builtin_prefetch(&e0[i + stride], 0, 0);     // -> global_prefetch_b8
      __builtin_prefetch(&e1[i + stride], 0, 0);
      __builtin_prefetch(&probs[i + stride], 0, 0);
    }
    const int r = cluster[e0[i]];
    const int c = cluster[e1[i]];
    const int key = r * n_nodes + c;            // < 2^31 for N = 40000
    const float p = probs[i];
    unsigned h = ((unsigned)key * 2654435761u) & smask;
    for (;;) {
      const int prev = atomicCAS(&hkeys[h], EMPTY_KEY, key);
      if (prev == EMPTY_KEY || prev == key) {
        atomicAdd(&hsum[h], p);
        atomicAdd(&hcnt[h], 1);
        break;
      }
      h = (h + 1) & smask;                      // linear probe (L2-resident)
    }
  }
}

// ---------------- Phase 3b: compact unique pairs + means -------------------
__global__ void k_compact(const int* __restrict__ hkeys, const float* __restrict__ hsum,
                          const int* __restrict__ hcnt, int slots,
                          float* __restrict__ o_row, float* __restrict__ o_col,
                          float* __restrict__ o_prob, int* __restrict__ counter,
                          int n_nodes) {
  const int stride = gridDim.x * blockDim.x;
  for (int s = blockIdx.x * blockDim.x + threadIdx.x; s < slots; s += stride) {
    const int cnt = hcnt[s];
    if (cnt > 0) {
      const int pos = atomicAdd(counter, 1);
      const unsigned key = (unsigned)hkeys[s];
      o_row[pos]  = (float)(key / (unsigned)n_nodes);
      o_col[pos]  = (float)(key % (unsigned)n_nodes);
      o_prob[pos] = hsum[s] / (float)cnt;
    }
  }
}

__global__ void k_final(const int* __restrict__ ws_nb, float* __restrict__ o_batch, int n) {
  int i = blockIdx.x * blockDim.x + threadIdx.x;
  if (i < n) o_batch[i] = (float)ws_nb[i];
}

// ---------------------------------------------------------------------------
extern "C" void kernel_launch(void* const* d_in, const int* in_sizes, int n_in,
                              void* d_out, int out_size, void* d_ws, size_t ws_size,
                              hipStream_t stream) {
  const int*   ei    = (const int*)d_in[0];     // [2, E]
  const float* ep    = (const float*)d_in[1];   // [E]
  const int*   cei   = (const int*)d_in[2];     // [2, NC]
  const float* ces   = (const float*)d_in[3];   // [NC]
  const int*   batch = (const int*)d_in[4];     // [N]

  const int E  = in_sizes[1];
  const int NC = in_sizes[3];
  const int N  = in_sizes[4];
  const int Ncl = (N > MAXN) ? MAXN : N;        // LDS-resident capacity guard

  const int* e0 = ei;
  const int* e1 = ei + (size_t)E;
  const int* c0 = cei;
  const int* c1 = cei + (size_t)NC;

  float* out     = (float*)d_out;               // tuple flattened, return order
  float* o_row   = out;
  float* o_col   = out + (size_t)E;
  float* o_prob  = out + (size_t)2 * E;
  float* o_clst  = out + (size_t)3 * E;
  float* o_score = o_clst + (size_t)N;
  float* o_batch = o_score + (size_t)NC;
  float* o_red   = o_batch + (size_t)N;

  // ---- workspace carve-up (deterministic in ws_size) ----
  char* w = (char*)d_ws;
  char* w0 = w;
  auto carve = [&](size_t bytes) -> char* {
    char* p = w; w += (bytes + 255) & ~(size_t)255; return p;
  };
  int* ws_cluster = (int*)carve((size_t)N * 4);
  int* ws_nb      = (int*)carve((size_t)N * 4);
  int* ws_counter = (int*)carve(256);
  const size_t fixed = (size_t)(w - w0);
  size_t slots = (size_t)1 << 24;               // 16M slots (0.5 load @ 8M edges)
  while (slots > ((size_t)1 << 20) && fixed + slots * 12 + 768 > ws_size) slots >>= 1;
  int*   hkeys = (int*)carve(slots * 4);
  float* hsum  = (float*)carve(slots * 4);
  int*   hcnt  = (int*)carve(slots * 4);
  const unsigned smask = (unsigned)slots - 1u;

  // 1. edge_score
  k_score<<<(NC + 255) / 256, 256, 0, stream>>>(ces, o_score, NC);

  // 2. init outputs + hash + misc
  {
    long long total = (long long)E;
    if ((long long)slots > total) total = (long long)slots;
    if ((long long)N > total)     total = (long long)N;
    const int blocks = (int)((total + 255) / 256);
    k_init<<<blocks, 256, 0, stream>>>(o_row, o_col, o_prob, E,
                                       hkeys, hsum, hcnt, (int)slots,
                                       ws_nb, N, ws_counter);
  }

  // 3. sequential greedy clustering (single wave32, LDS state + TDM feed)
  k_greedy<<<1, 32, 0, stream>>>(c0, c1, ces, ws_cluster, o_clst, o_red, Ncl, NC);

  // 4. new_batch scatter
  k_newbatch<<<(N + 255) / 256, 256, 0, stream>>>(ws_cluster, batch, ws_nb, N);

  // 5. hash-accumulate coalesce over 8M edges
  k_coalesce<<<4096, 256, 0, stream>>>(e0, e1, ep, ws_cluster,
                                       hkeys, hsum, hcnt, E, N, smask);

  // 6. compact unique pairs -> means
  k_compact<<<(int)((slots + 255) / 256), 256, 0, stream>>>(
      hkeys, hsum, hcnt, (int)slots, o_row, o_col, o_prob, ws_counter, N);

  // 7. finalize new_batch
  k_final<<<(N + 255) / 256, 256, 0, stream>>>(ws_nb, o_batch, N);
}